// PVM_Measurement_50216757625014
// MI455X (gfx1250) — compile-verified
//
#include <hip/hip_runtime.h>
#include <stdint.h>

// out[i,j] = (groups[i] == groups[j]) ? rho[i,j] : 0
// Pure memory-bound masking: ~1.05 GB traffic, ~45us floor at 23.3 TB/s.
// Strategy: b128 streaming with non-temporal hints for rho/out; stage the
// reused per-column group values into LDS via the gfx1250 async-copy path
// (global_load_async_to_lds_b128 + s_wait_asynccnt); 16-row blocking.

#define THREADS   256
#define TILE_COLS (THREADS * 4)   // 1024 columns per block
#define ROWS      16              // rows per block (11440 % 16 == 0)

typedef float f32x4 __attribute__((ext_vector_type(4)));
typedef int   i32x4 __attribute__((ext_vector_type(4)));

__global__ __launch_bounds__(THREADS)
void pvm_mask_kernel(const float* __restrict__ rho,
                     const int*   __restrict__ groups,
                     float*       __restrict__ out,
                     int d)
{
    __shared__ alignas(16) int s_gj[TILE_COLS]; // group of each column in tile
    __shared__ int s_gi[ROWS];                  // group of each row in tile

    const int tid = threadIdx.x;
    const int c4  = blockIdx.x * THREADS + tid; // float4-column index
    const int col = c4 * 4;
    const int i0  = blockIdx.y * ROWS;
    const int rmax = min(ROWS, d - i0);

    const bool vec_ok = ((d & 3) == 0) && (col + 3 < d);

    // ---- stage this lane's 4 column-group values into LDS via the CDNA5
    //      async copy pipe (ASYNCcnt). LDS byte offset = low 32 bits of the
    //      generic address of a __shared__ object (flat-LDS aperture). ----
    if (vec_ok) {
        const uint32_t lds_off = (uint32_t)(uintptr_t)(&s_gj[tid * 4]);
        const uint64_t gaddr   = (uint64_t)(uintptr_t)(groups + col);
        asm volatile("global_load_async_to_lds_b128 %0, %1, off"
                     :
                     : "v"(lds_off), "v"(gaddr)
                     : "memory");
    }
    // row-group values: one DS store per row, reused by all 1024 columns
    if (tid < ROWS && (i0 + tid) < d) {
        s_gi[tid] = groups[i0 + tid];
    }
    asm volatile("s_wait_asynccnt 0" ::: "memory");
    __syncthreads();

    if (vec_ok) {
        const i32x4 gj = *(const i32x4*)(&s_gj[tid * 4]);
        size_t base = (size_t)i0 * (size_t)d + (size_t)col;
        if (rmax == ROWS) {
#pragma unroll
            for (int r = 0; r < ROWS; ++r) {
                const int gi = s_gi[r];
                f32x4 v = __builtin_nontemporal_load((const f32x4*)(rho + base));
                f32x4 o;
                o.x = (gi == gj.x) ? v.x : 0.0f;
                o.y = (gi == gj.y) ? v.y : 0.0f;
                o.z = (gi == gj.z) ? v.z : 0.0f;
                o.w = (gi == gj.w) ? v.w : 0.0f;
                __builtin_nontemporal_store(o, (f32x4*)(out + base));
                base += (size_t)d;
            }
        } else {
            for (int r = 0; r < rmax; ++r) {
                const int gi = s_gi[r];
                f32x4 v = __builtin_nontemporal_load((const f32x4*)(rho + base));
                f32x4 o;
                o.x = (gi == gj.x) ? v.x : 0.0f;
                o.y = (gi == gj.y) ? v.y : 0.0f;
                o.z = (gi == gj.z) ? v.z : 0.0f;
                o.w = (gi == gj.w) ? v.w : 0.0f;
                __builtin_nontemporal_store(o, (f32x4*)(out + base));
                base += (size_t)d;
            }
        }
    } else {
        // scalar tail path (partial vec4 or d not divisible by 4)
        for (int k = 0; k < 4; ++k) {
            const int c = col + k;
            if (c >= d) break;
            const int gj = groups[c];
            size_t base = (size_t)i0 * (size_t)d + (size_t)c;
            for (int r = 0; r < rmax; ++r) {
                const int gi = s_gi[r];
                const float v = rho[base];
                out[base] = (gi == gj) ? v : 0.0f;
                base += (size_t)d;
            }
        }
    }
}

extern "C" void kernel_launch(void* const* d_in, const int* in_sizes, int n_in,
                              void* d_out, int out_size, void* d_ws, size_t ws_size,
                              hipStream_t stream)
{
    const float* rho    = (const float*)d_in[0];
    const int*   groups = (const int*)d_in[1];
    float*       out    = (float*)d_out;
    const int d = in_sizes[1];                     // 11440 for the reference

    const int c4_count = (d + 3) / 4;              // float4 columns
    dim3 block(THREADS);
    dim3 grid((c4_count + THREADS - 1) / THREADS,  // 12 column blocks
              (d + ROWS - 1) / ROWS);              // 715 row blocks
    pvm_mask_kernel<<<grid, block, 0, stream>>>(rho, groups, out, d);
}